// BidirectionalHMM_56109452755514
// MI455X (gfx1250) — compile-verified
//
#include <hip/hip_runtime.h>

#define T_LEN    4096
#define BATCH    32
#define OBS_D    128
#define NS       64
#define TB       (T_LEN * BATCH)              /* 131072 rows of the emission GEMM */
#define HALF_OUT ((long)TB * NS)              /* elements in one output half      */

typedef __attribute__((ext_vector_type(2))) float v2f;
typedef __attribute__((ext_vector_type(8))) float v8f;

// ---------------------------------------------------------------------------
// f32 16x16x4 WMMA (exact f32 path; no precision loss vs reference)
// ---------------------------------------------------------------------------
__device__ __forceinline__ v8f wmma_f32_16x16x4(v2f a, v2f b, v8f c) {
#if __has_builtin(__builtin_amdgcn_wmma_f32_16x16x4_f32)
  // 8 args: (neg_a, A, neg_b, B, c_mod, C, reuse_a, reuse_b)
  return __builtin_amdgcn_wmma_f32_16x16x4_f32(false, a, false, b, (short)0, c,
                                               false, false);
#else
  v8f d;
  asm volatile("v_wmma_f32_16x16x4_f32 %0, %1, %2, %3"
               : "=v"(d)
               : "v"(a), "v"(b), "v"(c));
  return d;
#endif
}

// ---------------------------------------------------------------------------
// Kernel 1: emissions GEMM  e[t,b,s] = obs[t,b,:] . emit_w[s,:] + emit_b[s]
// One wave -> one 16-row tile x all 64 states. 8 waves / block.
// Result written to BOTH halves of d_out (alpha region and beta region);
// the scan kernels consume it in place.
// ---------------------------------------------------------------------------
__global__ __launch_bounds__(256) void hmm_emit_gemm(
    const float* __restrict__ obs, const float* __restrict__ w,
    const float* __restrict__ bias, float* __restrict__ out) {
  const int lane  = threadIdx.x & 31;
  const int wave  = threadIdx.x >> 5;
  const int halfw = lane >> 4;          // 0 / 1
  const int m     = lane & 15;          // row (A) / col (B) within tile
  const int koff  = halfw * 2;          // K sub-offset held by this lane

  const long row0 = ((long)blockIdx.x * 8 + wave) * 16;

  const float* aptr = obs + (row0 + m) * OBS_D + koff;
  const float* w0   = w + (0 * 16 + m) * OBS_D + koff;   // B[k][n] = w[n][k]
  const float* w1   = w + (1 * 16 + m) * OBS_D + koff;
  const float* w2   = w + (2 * 16 + m) * OBS_D + koff;
  const float* w3   = w + (3 * 16 + m) * OBS_D + koff;

  v8f acc0 = {}, acc1 = {}, acc2 = {}, acc3 = {};
#pragma unroll
  for (int kk = 0; kk < OBS_D; kk += 4) {
    v2f a;  a.x  = aptr[kk]; a.y  = aptr[kk + 1];
    v2f b0; b0.x = w0[kk];   b0.y = w0[kk + 1];
    v2f b1; b1.x = w1[kk];   b1.y = w1[kk + 1];
    v2f b2; b2.x = w2[kk];   b2.y = w2[kk + 1];
    v2f b3; b3.x = w3[kk];   b3.y = w3[kk + 1];
    acc0 = wmma_f32_16x16x4(a, b0, acc0);
    acc1 = wmma_f32_16x16x4(a, b1, acc1);
    acc2 = wmma_f32_16x16x4(a, b2, acc2);
    acc3 = wmma_f32_16x16x4(a, b3, acc3);
  }

  const float bv0 = bias[0 * 16 + m];
  const float bv1 = bias[1 * 16 + m];
  const float bv2 = bias[2 * 16 + m];
  const float bv3 = bias[3 * 16 + m];

#pragma unroll
  for (int r = 0; r < 8; ++r) {
    const long rr   = row0 + r + 8 * halfw;
    const long base = rr * NS + m;
    const float o0 = acc0[r] + bv0;
    const float o1 = acc1[r] + bv1;
    const float o2 = acc2[r] + bv2;
    const float o3 = acc3[r] + bv3;
    out[base + 0]  = o0;  out[base + 16] = o1;
    out[base + 32] = o2;  out[base + 48] = o3;
    out[HALF_OUT + base + 0]  = o0;  out[HALF_OUT + base + 16] = o1;
    out[HALF_OUT + base + 32] = o2;  out[HALF_OUT + base + 48] = o3;
  }
}

// ---------------------------------------------------------------------------
// Kernel 2: forward/backward log-space scan, WMMA in the exp domain.
// grid = 4 single-wave blocks: (dir in {fwd,bwd}) x (batch tile in {0,1}).
//   h'[b,j] = e[b,j] + m_b + log( sum_i exp(h[b,i]-m_b) * exp(logT[i,j]) )
// The inner sum is a 16x64 @ 64x64 f32 matmul -> 64 v_wmma per step.
// ---------------------------------------------------------------------------
__global__ __launch_bounds__(32) void hmm_scan(
    const float* __restrict__ trans, float* __restrict__ out) {
  __shared__ float EL[NS][68];   // exp(log_softmax(trans)) (row- or col-wise)
  __shared__ float hb[16][68];   // current h, padded rows -> conflict-free
  __shared__ float pb[16][68];   // exp(h - m_b)
  __shared__ float mv[16];       // per-batch row max

  const int lane  = threadIdx.x;
  const int dir   = blockIdx.x >> 1;   // 0 = alpha (fwd), 1 = beta (bwd)
  const int b0    = (blockIdx.x & 1) * 16;
  float* half = out + (long)dir * HALF_OUT;

  // Build EL = exp(log_softmax). Fwd: softmax rows of trans.
  // Bwd: log_softmax(trans^T) -> EL[i][j] = softmax of column i at row j.
  for (int i = lane; i < NS; i += 32) {
    float mx = -1e30f;
    if (dir == 0) {
      for (int j = 0; j < NS; ++j) mx = fmaxf(mx, trans[i * NS + j]);
      float s = 0.f;
      for (int j = 0; j < NS; ++j) s += __expf(trans[i * NS + j] - mx);
      const float inv = 1.0f / s;
      for (int j = 0; j < NS; ++j) EL[i][j] = __expf(trans[i * NS + j] - mx) * inv;
    } else {
      for (int j = 0; j < NS; ++j) mx = fmaxf(mx, trans[j * NS + i]);
      float s = 0.f;
      for (int j = 0; j < NS; ++j) s += __expf(trans[j * NS + i] - mx);
      const float inv = 1.0f / s;
      for (int j = 0; j < NS; ++j) EL[i][j] = __expf(trans[j * NS + i] - mx) * inv;
    }
  }
  for (int idx = lane; idx < 16 * NS; idx += 32) hb[idx >> 6][idx & 63] = 0.f;
  __syncthreads();

  const int m     = lane & 15;
  const int halfw = lane >> 4;
  const int koff  = halfw * 2;

  for (int step = 0; step < T_LEN; ++step) {
    const int t = dir ? (T_LEN - 1 - step) : step;

    // (1) per-batch max of h
    if (lane < 16) {
      float mx = hb[lane][0];
      for (int j = 1; j < NS; ++j) mx = fmaxf(mx, hb[lane][j]);
      mv[lane] = mx;
    }
    __syncthreads();

    // (2) p = exp(h - m_b); lane handles batch m, 32 states per lane
    {
      const float mm   = mv[m];
      const int jbase  = halfw * 32;
#pragma unroll
      for (int j = 0; j < 32; ++j)
        pb[m][jbase + j] = __expf(hb[m][jbase + j] - mm);
    }
    __syncthreads();

    // (3) s = p @ EL via 64 chained f32 WMMAs
    v8f acc0 = {}, acc1 = {}, acc2 = {}, acc3 = {};
#pragma unroll
    for (int kk = 0; kk < NS; kk += 4) {
      v2f a;  a.x  = pb[m][kk + koff];          a.y  = pb[m][kk + koff + 1];
      v2f c0; c0.x = EL[kk + koff][0 * 16 + m]; c0.y = EL[kk + koff + 1][0 * 16 + m];
      v2f c1; c1.x = EL[kk + koff][1 * 16 + m]; c1.y = EL[kk + koff + 1][1 * 16 + m];
      v2f c2; c2.x = EL[kk + koff][2 * 16 + m]; c2.y = EL[kk + koff + 1][2 * 16 + m];
      v2f c3; c3.x = EL[kk + koff][3 * 16 + m]; c3.y = EL[kk + koff + 1][3 * 16 + m];
      acc0 = wmma_f32_16x16x4(a, c0, acc0);
      acc1 = wmma_f32_16x16x4(a, c1, acc1);
      acc2 = wmma_f32_16x16x4(a, c2, acc2);
      acc3 = wmma_f32_16x16x4(a, c3, acc3);
    }

    // prefetch next timestep's emission rows (global_prefetch_b8)
    {
      const int nt = dir ? (t - 1) : (t + 1);
      if (nt >= 0 && nt < T_LEN) {
        const long nb = (long)nt * (BATCH * NS) + (long)(b0 + m) * NS + halfw * 32;
        __builtin_prefetch(&half[nb], 0, 1);
      }
    }

    __syncthreads();

    // (4) h' = e + m_b + log(s); write to output and to hb for next step
    const long tbase = (long)t * (BATCH * NS);
#pragma unroll
    for (int r = 0; r < 8; ++r) {
      const int  bl   = r + 8 * halfw;
      const float mmb = mv[bl];
      const long base = tbase + (long)(b0 + bl) * NS + m;
      const float h0 = half[base + 0]  + mmb + __logf(acc0[r]);
      const float h1 = half[base + 16] + mmb + __logf(acc1[r]);
      const float h2 = half[base + 32] + mmb + __logf(acc2[r]);
      const float h3 = half[base + 48] + mmb + __logf(acc3[r]);
      half[base + 0]  = h0;  half[base + 16] = h1;
      half[base + 32] = h2;  half[base + 48] = h3;
      hb[bl][0 * 16 + m] = h0;  hb[bl][1 * 16 + m] = h1;
      hb[bl][2 * 16 + m] = h2;  hb[bl][3 * 16 + m] = h3;
    }
    __syncthreads();
  }
}

// ---------------------------------------------------------------------------
extern "C" void kernel_launch(void* const* d_in, const int* in_sizes, int n_in,
                              void* d_out, int out_size, void* d_ws,
                              size_t ws_size, hipStream_t stream) {
  (void)in_sizes; (void)n_in; (void)out_size; (void)d_ws; (void)ws_size;
  const float* obs   = (const float*)d_in[0];
  const float* w     = (const float*)d_in[1];
  const float* bias  = (const float*)d_in[2];
  const float* trans = (const float*)d_in[3];
  float* out = (float*)d_out;

  // Phase 1: emissions into both output halves (8192 row tiles, 8 waves/block)
  hmm_emit_gemm<<<TB / 16 / 8, 256, 0, stream>>>(obs, w, bias, out);
  // Phase 2: 4 independent single-wave scan chains (2 dirs x 2 batch tiles)
  hmm_scan<<<4, 32, 0, stream>>>(trans, out);
}